// CausalDiscoverer_18940805776184
// MI455X (gfx1250) — compile-verified
//
#include <hip/hip_runtime.h>
#include <math.h>

// ---------------------------------------------------------------------------
// CausalDiscoverer forward for MI455X (gfx1250, wave32).
//   L1: scalar message passing + scalar->64 MLPs (VALU)
//   L2: 64-dim message passing (f32 atomics) + MLPs via f32 WMMA 16x16x4
//   emlp: factorized concat-linear -> A[i]+B[j], then 1M-pair LN/GELU/dot
// ---------------------------------------------------------------------------

#define KN   1024
#define KDIM 64
#define KEC  262144
#define KEI  32768
#define KEPS 1e-5f

typedef __attribute__((ext_vector_type(2))) float v2f;
typedef __attribute__((ext_vector_type(8))) float v8f;

__device__ __forceinline__ float gelu_exact(float x) {
  return 0.5f * x * (1.0f + erff(x * 0.7071067811865475f));
}

// ------------------------- layer-1 edge kernels ----------------------------
__global__ void edge1_corr_k(const float* __restrict__ x, const int* __restrict__ ei,
                             const float* __restrict__ ea, const float* __restrict__ lw,
                             const float* __restrict__ lb, float* __restrict__ agg) {
  int e = blockIdx.x * 256 + threadIdx.x;
  if (e < KEC) {
    int s = ei[e], d = ei[KEC + e];
    float m = x[s] + ea[e] * lw[0] + lb[0];
    m = fmaxf(m, 0.0f);
    atomicAdd(&agg[d], m);
  }
}

__global__ void edge1_impl_k(const float* __restrict__ x, const int* __restrict__ ei,
                             float* __restrict__ agg) {
  int e = blockIdx.x * 256 + threadIdx.x;
  if (e < KEI) {
    int s = ei[e], d = ei[KEI + e];
    atomicAdd(&agg[d], x[s]);
  }
}

// ------------------------- layer-2 edge kernels ----------------------------
__global__ void edge2_corr_k(const float* __restrict__ h, const int* __restrict__ ei,
                             const float* __restrict__ ea, const float* __restrict__ lw,
                             const float* __restrict__ lb, float* __restrict__ agg) {
  int tid = blockIdx.x * 256 + threadIdx.x;   // KEC*64 total
  int e = tid >> 6, f = tid & 63;
  if (e < KEC) {
    int s = ei[e], d = ei[KEC + e];
    float m = h[s * KDIM + f] + ea[e] * lw[f] + lb[f];
    m = fmaxf(m, 0.0f);
    atomicAdd(&agg[d * KDIM + f], m);
  }
}

__global__ void edge2_impl_k(const float* __restrict__ h, const int* __restrict__ ei,
                             float* __restrict__ agg) {
  int tid = blockIdx.x * 256 + threadIdx.x;   // KEI*64 total
  int e = tid >> 6, f = tid & 63;
  if (e < KEI) {
    int s = ei[e], d = ei[KEI + e];
    atomicAdd(&agg[d * KDIM + f], h[s * KDIM + f]);
  }
}

// ---- layer-1 node: u = s*W1 + b1 ; LayerNorm ; GELU  (both branches) ------
// block (32,8): one wave per row, lane handles features f and f+32
__global__ void l1_node_k(const float* __restrict__ x,
                          const float* __restrict__ aggc, const float* __restrict__ aggi,
                          const float* __restrict__ W1c, const float* __restrict__ b1c,
                          const float* __restrict__ gc,  const float* __restrict__ btc,
                          const float* __restrict__ W1i, const float* __restrict__ b1i,
                          const float* __restrict__ gi,  const float* __restrict__ bti,
                          float* __restrict__ t1, float* __restrict__ t2) {
  int lane = threadIdx.x;
  int row  = blockIdx.x * 8 + threadIdx.y;
  float sc = x[row] + aggc[row];
  float si = x[row] + aggi[row];
  int f0 = lane, f1 = lane + 32;
  float uc0 = sc * W1c[f0] + b1c[f0], uc1 = sc * W1c[f1] + b1c[f1];
  float ui0 = si * W1i[f0] + b1i[f0], ui1 = si * W1i[f1] + b1i[f1];
  float mc = uc0 + uc1, mi = ui0 + ui1;
  #pragma unroll
  for (int o = 16; o > 0; o >>= 1) { mc += __shfl_xor(mc, o); mi += __shfl_xor(mi, o); }
  mc *= (1.0f / 64.0f); mi *= (1.0f / 64.0f);
  float dc0 = uc0 - mc, dc1 = uc1 - mc, di0 = ui0 - mi, di1 = ui1 - mi;
  float vc = dc0 * dc0 + dc1 * dc1, vi = di0 * di0 + di1 * di1;
  #pragma unroll
  for (int o = 16; o > 0; o >>= 1) { vc += __shfl_xor(vc, o); vi += __shfl_xor(vi, o); }
  vc *= (1.0f / 64.0f); vi *= (1.0f / 64.0f);
  float ic = 1.0f / sqrtf(vc + KEPS), ii = 1.0f / sqrtf(vi + KEPS);
  t1[row * KDIM + f0] = gelu_exact(dc0 * ic * gc[f0] + btc[f0]);
  t1[row * KDIM + f1] = gelu_exact(dc1 * ic * gc[f1] + btc[f1]);
  t2[row * KDIM + f0] = gelu_exact(di0 * ii * gi[f0] + bti[f0]);
  t2[row * KDIM + f1] = gelu_exact(di1 * ii * gi[f1] + bti[f1]);
}

// ---------------- LayerNorm + GELU over rows of T (in place) ---------------
__global__ void lngelu_k(float* __restrict__ T, const float* __restrict__ g,
                         const float* __restrict__ bt) {
  int lane = threadIdx.x;
  int row  = blockIdx.x * 8 + threadIdx.y;
  int f0 = lane, f1 = lane + 32;
  float u0 = T[row * KDIM + f0], u1 = T[row * KDIM + f1];
  float m = u0 + u1;
  #pragma unroll
  for (int o = 16; o > 0; o >>= 1) m += __shfl_xor(m, o);
  m *= (1.0f / 64.0f);
  float d0 = u0 - m, d1 = u1 - m;
  float v = d0 * d0 + d1 * d1;
  #pragma unroll
  for (int o = 16; o > 0; o >>= 1) v += __shfl_xor(v, o);
  v *= (1.0f / 64.0f);
  float inv = 1.0f / sqrtf(v + KEPS);
  T[row * KDIM + f0] = gelu_exact(d0 * inv * g[f0] + bt[f0]);
  T[row * KDIM + f1] = gelu_exact(d1 * inv * g[f1] + bt[f1]);
}

// -------- Y[N,64] = (X (+Xadd)) @ W[64,64] + bias (+= Y if acc) ------------
// Full-f32 WMMA: V_WMMA_F32_16X16X4_F32 chained over K=64 (16 steps).
// Block = 128 threads (4 waves); wave w computes cols [16w,16w+16) of 16 rows.
__global__ void gemm64_wmma_k(const float* __restrict__ X, const float* __restrict__ Xadd,
                              const float* __restrict__ W, const float* __restrict__ bias,
                              float* __restrict__ Y, int accumulate) {
  __shared__ float sX[16 * 64];
  int tid = threadIdx.x;
  int m0  = blockIdx.x * 16;
  #pragma unroll
  for (int t = 0; t < 8; ++t) {
    int li = tid + t * 128;
    int r = li >> 6, c = li & 63;
    float v = X[(m0 + r) * KDIM + c];
    if (Xadd) v += Xadd[(m0 + r) * KDIM + c];
    sX[li] = v;
  }
  __syncthreads();
  int wave = tid >> 5;
  int lane = tid & 31;
  int half = lane >> 4;        // 0: lanes 0-15, 1: lanes 16-31
  int lrow = lane & 15;
  int n0   = wave * 16;
  v8f acc = {};
  #pragma unroll
  for (int ks = 0; ks < 16; ++ks) {
    int k0 = ks * 4;
    v2f a, b;
    // A 16x4 f32: lanes0-15 -> K={k0,k0+1}; lanes16-31 -> K={k0+2,k0+3}
    a.x = sX[lrow * 64 + k0 + 2 * half];
    a.y = sX[lrow * 64 + k0 + 2 * half + 1];
    // B 4x16 f32: v0 = rows {k0 (lo-half), k0+2 (hi-half)}, v1 = rows {k0+1, k0+3}
    b.x = W[(k0 + 2 * half) * KDIM + n0 + lrow];
    b.y = W[(k0 + 2 * half + 1) * KDIM + n0 + lrow];
    acc = __builtin_amdgcn_wmma_f32_16x16x4_f32(false, a, false, b, (short)0, acc,
                                                false, false);
  }
  int col = n0 + lrow;
  float bv = bias ? bias[col] : 0.0f;
  #pragma unroll
  for (int r = 0; r < 8; ++r) {
    int row = m0 + r + 8 * half;       // C/D: vgpr r -> M=r (lo) / M=8+r (hi)
    int idx = row * KDIM + col;
    float v = acc[r] + bv;
    if (accumulate) v += Y[idx];
    Y[idx] = v;
  }
}

// ------------------------- transpose N x 64 -> 64 x N ----------------------
__global__ void transpose_k(const float* __restrict__ B, float* __restrict__ Bt) {
  int tid = blockIdx.x * 256 + threadIdx.x;   // N*64 total
  int j = tid >> 6, k = tid & 63;
  Bt[k * KN + j] = B[tid];
}

// -------- all-pairs: out[i,j] = sigmoid( w2 . gelu(LN(A[i]+Bt[:,j])) + b2 ) -
__global__ void pair_k(const float* __restrict__ A, const float* __restrict__ Bt,
                       const float* __restrict__ g, const float* __restrict__ bt,
                       const float* __restrict__ w2, const float* __restrict__ b2,
                       float* __restrict__ out) {
  __shared__ float sA[64], sG[64], sB[64], sW[64];
  int i = blockIdx.y;
  int j = blockIdx.x * 256 + threadIdx.x;
  if (threadIdx.x < 64) {
    sA[threadIdx.x] = A[i * KDIM + threadIdx.x];
    sG[threadIdx.x] = g[threadIdx.x];
    sB[threadIdx.x] = bt[threadIdx.x];
    sW[threadIdx.x] = w2[threadIdx.x];
  }
  __syncthreads();
  float p[64];
  float s = 0.0f;
  #pragma unroll
  for (int k = 0; k < 64; ++k) { float v = sA[k] + Bt[k * KN + j]; p[k] = v; s += v; }
  float m = s * (1.0f / 64.0f);
  float var = 0.0f;
  #pragma unroll
  for (int k = 0; k < 64; ++k) { float d = p[k] - m; var += d * d; }
  var *= (1.0f / 64.0f);
  float inv = 1.0f / sqrtf(var + KEPS);
  float dot = 0.0f;
  #pragma unroll
  for (int k = 0; k < 64; ++k) {
    float gn = (p[k] - m) * inv * sG[k] + sB[k];
    dot += gelu_exact(gn) * sW[k];
  }
  float e = dot + b2[0];
  out[i * KN + j] = 1.0f / (1.0f + expf(-e));
}

// ---------------------------------------------------------------------------
extern "C" void kernel_launch(void* const* d_in, const int* in_sizes, int n_in,
                              void* d_out, int out_size, void* d_ws, size_t ws_size,
                              hipStream_t stream) {
  (void)n_in; (void)out_size; (void)ws_size;
  auto F = [&](int i) { return (const float*)d_in[i]; };
  auto I = [&](int i) { return (const int*)d_in[i]; };

  const float *x, *ea_corr; const int *ei_corr, *ei_impl;
  const float *lin1_w, *lin1_b, *lin2_w, *lin2_b;
  const float *e1W1, *e1b1, *e1g, *e1bt, *e1W2, *e1b2;   // gine1
  const float *i1W1, *i1b1, *i1g, *i1bt, *i1W2, *i1b2;   // gin1
  const float *e2W1, *e2b1, *e2g, *e2bt, *e2W2, *e2b2;   // gine2
  const float *i2W1, *i2b1, *i2g, *i2bt, *i2W2, *i2b2;   // gin2
  const float *mW1, *mb1, *mg, *mbt, *mW2, *mb2;         // emlp

  if (in_sizes[0] == KN) {
    // recursive insertion-order flattening of setup_inputs()
    x = F(0); ei_corr = I(1); ea_corr = F(2); ei_impl = I(3);
    lin1_w = F(4); lin1_b = F(5);
    e1W1 = F(6);  e1b1 = F(7);  e1g = F(8);  e1bt = F(9);  e1W2 = F(10); e1b2 = F(11);
    i1W1 = F(12); i1b1 = F(13); i1g = F(14); i1bt = F(15); i1W2 = F(16); i1b2 = F(17);
    lin2_w = F(18); lin2_b = F(19);
    e2W1 = F(20); e2b1 = F(21); e2g = F(22); e2bt = F(23); e2W2 = F(24); e2b2 = F(25);
    i2W1 = F(26); i2b1 = F(27); i2g = F(28); i2bt = F(29); i2W2 = F(30); i2b2 = F(31);
    mW1 = F(32); mb1 = F(33); mg = F(34); mbt = F(35); mW2 = F(36); mb2 = F(37);
  } else {
    // jax.tree_util.tree_flatten order (dict keys sorted at every level)
    ea_corr = F(0); ei_corr = I(1); ei_impl = I(2);
    mW1 = F(3); mW2 = F(4); mb1 = F(5); mb2 = F(6); mbt = F(7); mg = F(8);
    i1W1 = F(9);  i1W2 = F(10); i1b1 = F(11); i1b2 = F(12); i1bt = F(13); i1g = F(14);
    i2W1 = F(15); i2W2 = F(16); i2b1 = F(17); i2b2 = F(18); i2bt = F(19); i2g = F(20);
    e1W1 = F(21); e1W2 = F(22); e1b1 = F(23); e1b2 = F(24); e1bt = F(25); e1g = F(26);
    e2W1 = F(27); e2W2 = F(28); e2b1 = F(29); e2b2 = F(30); e2bt = F(31); e2g = F(32);
    lin1_b = F(33); lin1_w = F(34); lin2_b = F(35); lin2_w = F(36);
    x = F(37);
  }

  // ---- workspace layout (floats) ----
  float* ws = (float*)d_ws;
  float* aggc1 = ws;                       // KN
  float* aggi1 = aggc1 + KN;               // KN
  float* aggc2 = aggi1 + KN;               // KN*KDIM
  float* aggi2 = aggc2 + KN * KDIM;        // KN*KDIM
  float* t1    = aggi2 + KN * KDIM;        // KN*KDIM
  float* t2    = t1    + KN * KDIM;
  float* h1    = t2    + KN * KDIM;
  float* h2    = h1    + KN * KDIM;
  float* eA    = h2    + KN * KDIM;
  float* eB    = eA    + KN * KDIM;
  float* eBt   = eB    + KN * KDIM;

  // zero the atomic-accumulation region (aggc1..aggi2 contiguous)
  (void)hipMemsetAsync(d_ws, 0, (size_t)(2 * KN + 2 * KN * KDIM) * sizeof(float), stream);

  dim3 waveRows(32, 8);

  // ---------------- layer 1 ----------------
  edge1_corr_k<<<KEC / 256, 256, 0, stream>>>(x, ei_corr, ea_corr, lin1_w, lin1_b, aggc1);
  edge1_impl_k<<<KEI / 256, 256, 0, stream>>>(x, ei_impl, aggi1);
  l1_node_k<<<KN / 8, waveRows, 0, stream>>>(x, aggc1, aggi1,
                                             e1W1, e1b1, e1g, e1bt,
                                             i1W1, i1b1, i1g, i1bt, t1, t2);
  gemm64_wmma_k<<<KN / 16, 128, 0, stream>>>(t1, nullptr, e1W2, e1b2, h1, 0);
  gemm64_wmma_k<<<KN / 16, 128, 0, stream>>>(t2, nullptr, i1W2, i1b2, h1, 1);

  // ---------------- layer 2 ----------------
  edge2_corr_k<<<KEC * KDIM / 256, 256, 0, stream>>>(h1, ei_corr, ea_corr, lin2_w, lin2_b, aggc2);
  edge2_impl_k<<<KEI * KDIM / 256, 256, 0, stream>>>(h1, ei_impl, aggi2);
  gemm64_wmma_k<<<KN / 16, 128, 0, stream>>>(h1, aggc2, e2W1, e2b1, t1, 0);
  lngelu_k<<<KN / 8, waveRows, 0, stream>>>(t1, e2g, e2bt);
  gemm64_wmma_k<<<KN / 16, 128, 0, stream>>>(t1, nullptr, e2W2, e2b2, h2, 0);
  gemm64_wmma_k<<<KN / 16, 128, 0, stream>>>(h1, aggi2, i2W1, i2b1, t2, 0);
  lngelu_k<<<KN / 8, waveRows, 0, stream>>>(t2, i2g, i2bt);
  gemm64_wmma_k<<<KN / 16, 128, 0, stream>>>(t2, nullptr, i2W2, i2b2, h2, 1);

  // ---------------- all-pairs edge MLP (factorized concat) ----------------
  gemm64_wmma_k<<<KN / 16, 128, 0, stream>>>(h2, nullptr, mW1,        mb1,     eA, 0);
  gemm64_wmma_k<<<KN / 16, 128, 0, stream>>>(h2, nullptr, mW1 + 4096, nullptr, eB, 0);
  transpose_k<<<KN * KDIM / 256, 256, 0, stream>>>(eB, eBt);
  dim3 pgrid(KN / 256, KN);
  pair_k<<<pgrid, 256, 0, stream>>>(eA, eBt, mg, mbt, mW2, mb2, (float*)d_out);
}